// HeteroGNN_1288490189190
// MI455X (gfx1250) — compile-verified
//
#include <hip/hip_runtime.h>

typedef float v2f __attribute__((ext_vector_type(2)));
typedef float v8f __attribute__((ext_vector_type(8)));

#define HF 128  // hidden width

// ---------------- elementwise / utility kernels ----------------

__global__ void fill_f32(float* __restrict__ p, float v, size_t n) {
  size_t i = (size_t)blockIdx.x * blockDim.x + threadIdx.x;
  size_t stride = (size_t)gridDim.x * blockDim.x;
  for (; i < n; i += stride) p[i] = v;
}

__global__ void count_edges(const int* __restrict__ dst, int ne, float* __restrict__ cnt) {
  int e = blockIdx.x * blockDim.x + threadIdx.x;
  if (e < ne) unsafeAtomicAdd(&cnt[dst[e]], 1.0f);
}

__global__ void recip_clamp(float* __restrict__ c, int n) {
  int i = blockIdx.x * blockDim.x + threadIdx.x;
  if (i < n) c[i] = 1.0f / fmaxf(c[i], 1.0f);
}

// one wave per edge; 32 lanes x float4 = 128 features.
// gfx1250 prefetch of a look-ahead edge's source row hides gather latency.
#define EDGE_AHEAD 32
__global__ void scatter_add_h128(const float* __restrict__ x, const int* __restrict__ src,
                                 const int* __restrict__ dst, int ne, float* __restrict__ s) {
  int e = blockIdx.x * (blockDim.x >> 5) + (threadIdx.x >> 5);
  if (e >= ne) return;
  int lane = threadIdx.x & 31;
  int ep = e + EDGE_AHEAD;
  if (ep < ne) {
    // wave-uniform index load, per-row prefetch into caches (global_prefetch_b8)
    __builtin_prefetch(x + (size_t)src[ep] * HF + lane * 4, 0, 3);
  }
  size_t so  = (size_t)src[e] * HF + lane * 4;
  size_t so2 = (size_t)dst[e] * HF + lane * 4;
  float4 v = *(const float4*)(x + so);
  unsafeAtomicAdd(&s[so2 + 0], v.x);
  unsafeAtomicAdd(&s[so2 + 1], v.y);
  unsafeAtomicAdd(&s[so2 + 2], v.z);
  unsafeAtomicAdd(&s[so2 + 3], v.w);
}

// ---------------- fp32 WMMA GEMM: C[M x N] (+)= (A .* rowscale)[M x K] @ W[K x N] (+ bias) ------
// block = (32, N/16); one 32-row M-tile staged in padded LDS, shared by all waves.
// Each wave computes a 32x16 output tile: two independent WMMA chains share one B fragment
// per k-step (halves B traffic, doubles matrix-op ILP). Rows are clamped, never predicated,
// so the k-loop runs with full EXEC (WMMA requirement).

__global__ void gemm_f32_wmma(const float* __restrict__ A, const float* __restrict__ W,
                              const float* __restrict__ bias, const float* __restrict__ rowscale,
                              float* __restrict__ C, int M, int K, int N, int accum)
{
  extern __shared__ float As[];                 // 32 x (K+4)
  const int lane = threadIdx.x;                 // 0..31
  const int wv   = threadIdx.y;                 // N-tile index
  const int tm   = blockIdx.x;                  // 32-row M tile
  const int nth  = 32 * blockDim.y;
  const int tid  = wv * 32 + lane;
  const int Kp   = K + 4;                       // LDS row stride (floats)
  const int base = tm * 32;

  // ---- cooperative, coalesced fill of 32-row A tile; rowscale folded in; rows clamped ----
  const int cpr = K >> 2;                       // float4 chunks per row (16/32/64)
  const int shf = 31 - __clz(cpr);              // log2(cpr)
  for (int i = tid; i < (cpr << 5); i += nth) {
    int row = i >> shf;
    int ch  = (i & (cpr - 1)) << 2;
    int rg  = base + row; if (rg > M - 1) rg = M - 1;
    float rs = rowscale ? rowscale[rg] : 1.0f;
    float4 v = *(const float4*)(A + (size_t)rg * K + ch);
    float4 o; o.x = v.x * rs; o.y = v.y * rs; o.z = v.z * rs; o.w = v.w * rs;
    *(float4*)(As + row * Kp + ch) = o;         // 16B-aligned ds_store_b128
  }
  __syncthreads();

  const int half = lane >> 4;                   // 0 or 1
  const int l16  = lane & 15;
  const int colB = wv * 16 + l16;

  v8f acc0 = {0.f, 0.f, 0.f, 0.f, 0.f, 0.f, 0.f, 0.f};
  v8f acc1 = {0.f, 0.f, 0.f, 0.f, 0.f, 0.f, 0.f, 0.f};
  if (accum) {
#pragma unroll
    for (int r = 0; r < 8; ++r) {
      int row0 = base + r + 8 * half;           // C/D layout: VGPR r -> M = r + 8*half
      int row1 = row0 + 16;
      if (row0 > M - 1) row0 = M - 1;           // clamped (never stored for padded rows)
      if (row1 > M - 1) row1 = M - 1;
      acc0[r] = C[(size_t)row0 * N + colB];
      acc1[r] = C[(size_t)row1 * N + colB];
    }
  }

  const size_t wstep = (size_t)4 * N;
  const float* wp0 = W + (size_t)(half * 2) * N + colB;  // B row k0
  const float* wp1 = wp0 + N;                            // B row k0+1
  const float* ap0 = As + l16 * Kp + half * 2;           // A frag rows 0..15
  const float* ap1 = ap0 + 16 * Kp;                      // A frag rows 16..31

#pragma unroll 4
  for (int k = 0; k < K; k += 4) {
    v2f b; b.x = *wp0; b.y = *wp1;
    v2f a0 = *(const v2f*)ap0;
    v2f a1 = *(const v2f*)ap1;
    acc0 = __builtin_amdgcn_wmma_f32_16x16x4_f32(false, a0, false, b, (short)0, acc0, false, false);
    acc1 = __builtin_amdgcn_wmma_f32_16x16x4_f32(false, a1, false, b, (short)0, acc1, false, false);
    ap0 += 4; ap1 += 4;
    wp0 += wstep; wp1 += wstep;
  }

  const float bv = bias ? bias[colB] : 0.0f;
#pragma unroll
  for (int r = 0; r < 8; ++r) {
    int row0 = base + r + 8 * half;
    int row1 = row0 + 16;
    if (row0 < M) C[(size_t)row0 * N + colB] = acc0[r] + bv;
    if (row1 < M) C[(size_t)row1 * N + colB] = acc1[r] + bv;
  }
}

// ---------------- BatchNorm (training-mode batch stats, biased var) ----------------

__global__ void col_stats_h128(const float* __restrict__ X, int M,
                               float* __restrict__ mean, float* __restrict__ inv) {
  const int h = blockIdx.x;                     // one block per feature column
  float s = 0.f, s2 = 0.f;
  for (int i = threadIdx.x; i < M; i += blockDim.x) {
    float v = X[(size_t)i * HF + h];
    s += v; s2 += v * v;
  }
  __shared__ float sh[256], sh2[256];
  sh[threadIdx.x] = s; sh2[threadIdx.x] = s2;
  __syncthreads();
  for (int st = blockDim.x >> 1; st > 0; st >>= 1) {
    if ((int)threadIdx.x < st) {
      sh[threadIdx.x]  += sh[threadIdx.x + st];
      sh2[threadIdx.x] += sh2[threadIdx.x + st];
    }
    __syncthreads();
  }
  if (threadIdx.x == 0) {
    float mu  = sh[0] / (float)M;
    float var = sh2[0] / (float)M - mu * mu;
    mean[h] = mu;
    inv[h]  = rsqrtf(var + 1e-5f);
  }
}

// o = 0.5*(o + bl1 + bl2)  (per-column biases)
__global__ void finish_o(float* __restrict__ o, const float* __restrict__ b1,
                         const float* __restrict__ b2, size_t n) {
  size_t i = (size_t)blockIdx.x * blockDim.x + threadIdx.x;
  size_t stride = (size_t)gridDim.x * blockDim.x;
  for (; i < n; i += stride) {
    int h = (int)(i & (HF - 1));
    o[i] = 0.5f * (o[i] + b1[h] + b2[h]);
  }
}

__global__ void bn_relu_inplace(float* __restrict__ X, size_t n,
                                const float* __restrict__ mean, const float* __restrict__ inv,
                                const float* __restrict__ g, const float* __restrict__ b) {
  size_t i = (size_t)blockIdx.x * blockDim.x + threadIdx.x;
  size_t stride = (size_t)gridDim.x * blockDim.x;
  for (; i < n; i += stride) {
    int h = (int)(i & (HF - 1));
    float v = (X[i] - mean[h]) * inv[h] * g[h] + b[h];
    X[i] = fmaxf(v, 0.0f);
  }
}

// A = relu(bn(o)) + A  (residual)
__global__ void bn_relu_residual(const float* __restrict__ o, float* __restrict__ A, size_t n,
                                 const float* __restrict__ mean, const float* __restrict__ inv,
                                 const float* __restrict__ g, const float* __restrict__ b) {
  size_t i = (size_t)blockIdx.x * blockDim.x + threadIdx.x;
  size_t stride = (size_t)gridDim.x * blockDim.x;
  for (; i < n; i += stride) {
    int h = (int)(i & (HF - 1));
    float v = (o[i] - mean[h]) * inv[h] * g[h] + b[h];
    A[i] = fmaxf(v, 0.0f) + A[i];
  }
}

// ---------------- host orchestration ----------------

static inline void gemm_launch(hipStream_t st, const float* A, const float* W,
                               const float* bias, const float* rowscale, float* C,
                               int M, int K, int N, int accum) {
  dim3 blk(32, N / 16);                          // 8 waves for N=128, 4 for N=64
  dim3 grd((M + 31) / 32);                       // 32-row M tiles
  size_t shmem = (size_t)32 * (K + 4) * sizeof(float);
  gemm_f32_wmma<<<grd, blk, shmem, st>>>(A, W, bias, rowscale, C, M, K, N, accum);
}

extern "C" void kernel_launch(void* const* d_in, const int* in_sizes, int n_in,
                              void* d_out, int out_size, void* d_ws, size_t ws_size,
                              hipStream_t stream) {
  (void)n_in; (void)out_size; (void)ws_size;
  const int ND = 30000, NG = 60000, NS = 15000;
  const int FD = 256, FG = 128, FS = 64, HH = HF, OO = 64;

  const float* x_drug = (const float*)d_in[0];
  const float* x_gene = (const float*)d_in[1];
  const float* x_dis  = (const float*)d_in[2];
  const int* src_dg = (const int*)d_in[3];  const int* dst_dg = (const int*)d_in[4];
  const int* src_gd = (const int*)d_in[5];  const int* dst_gd = (const int*)d_in[6];
  const int* src_gs = (const int*)d_in[7];  const int* dst_gs = (const int*)d_in[8];
  const int* src_sg = (const int*)d_in[9];  const int* dst_sg = (const int*)d_in[10];
  const int* src_ds = (const int*)d_in[11]; const int* dst_ds = (const int*)d_in[12];
  const int* src_sd = (const int*)d_in[13]; const int* dst_sd = (const int*)d_in[14];
  const float* W_in_drug = (const float*)d_in[15]; const float* b_in_drug = (const float*)d_in[16];
  const float* W_in_gene = (const float*)d_in[17]; const float* b_in_gene = (const float*)d_in[18];
  const float* W_in_dis  = (const float*)d_in[19]; const float* b_in_dis  = (const float*)d_in[20];
  const float* g_in  = (const float*)d_in[21]; const float* bt_in = (const float*)d_in[22];
  const float* Wl = (const float*)d_in[23];
  const float* Wr = (const float*)d_in[24];
  const float* bl = (const float*)d_in[25];
  const float* g_conv = (const float*)d_in[26]; const float* bt_conv = (const float*)d_in[27];
  const float* W_fin = (const float*)d_in[28]; const float* b_fin = (const float*)d_in[29];

  const int E_dg = in_sizes[3], E_gd = in_sizes[5], E_gs = in_sizes[7],
            E_sg = in_sizes[9], E_ds = in_sizes[11], E_sd = in_sizes[13];

  // workspace layout (floats)
  float* ws = (float*)d_ws;
  size_t off = 0;
  float* A_d = ws + off; off += (size_t)ND * HH;   // current features
  float* A_g = ws + off; off += (size_t)NG * HH;
  float* A_s = ws + off; off += (size_t)NS * HH;
  float* o_d = ws + off; off += (size_t)ND * HH;   // conv accumulators
  float* o_g = ws + off; off += (size_t)NG * HH;
  float* o_s = ws + off; off += (size_t)NS * HH;
  float* sb  = ws + off; off += (size_t)NG * HH;   // scatter buffer (max dst count)
  float* r_dg = ws + off; off += NG;               // 1/max(count,1), contiguous region
  float* r_gd = ws + off; off += ND;
  float* r_gs = ws + off; off += NS;
  float* r_sg = ws + off; off += NG;
  float* r_ds = ws + off; off += NS;
  float* r_sd = ws + off; off += ND;
  float* mean = ws + off; off += HH;
  float* inv  = ws + off; off += HH;

  const int TB = 256;
  const int GRD = 2048;

  // ---- edge counts (layer-invariant) -> reciprocal means ----
  int ncnt = NG + ND + NS + NG + NS + ND;
  fill_f32<<<GRD, TB, 0, stream>>>(r_dg, 0.f, (size_t)ncnt);
  count_edges<<<(E_dg + TB - 1) / TB, TB, 0, stream>>>(dst_dg, E_dg, r_dg);
  count_edges<<<(E_gd + TB - 1) / TB, TB, 0, stream>>>(dst_gd, E_gd, r_gd);
  count_edges<<<(E_gs + TB - 1) / TB, TB, 0, stream>>>(dst_gs, E_gs, r_gs);
  count_edges<<<(E_sg + TB - 1) / TB, TB, 0, stream>>>(dst_sg, E_sg, r_sg);
  count_edges<<<(E_ds + TB - 1) / TB, TB, 0, stream>>>(dst_ds, E_ds, r_ds);
  count_edges<<<(E_sd + TB - 1) / TB, TB, 0, stream>>>(dst_sd, E_sd, r_sd);
  recip_clamp<<<(ncnt + TB - 1) / TB, TB, 0, stream>>>(r_dg, ncnt);

  // ---- input projections + BN + ReLU ----
  gemm_launch(stream, x_drug, W_in_drug, b_in_drug, nullptr, A_d, ND, FD, HH, 0);
  col_stats_h128<<<HH, 256, 0, stream>>>(A_d, ND, mean, inv);
  bn_relu_inplace<<<GRD, TB, 0, stream>>>(A_d, (size_t)ND * HH, mean, inv, g_in + 0 * HH, bt_in + 0 * HH);

  gemm_launch(stream, x_gene, W_in_gene, b_in_gene, nullptr, A_g, NG, FG, HH, 0);
  col_stats_h128<<<HH, 256, 0, stream>>>(A_g, NG, mean, inv);
  bn_relu_inplace<<<GRD, TB, 0, stream>>>(A_g, (size_t)NG * HH, mean, inv, g_in + 1 * HH, bt_in + 1 * HH);

  gemm_launch(stream, x_dis, W_in_dis, b_in_dis, nullptr, A_s, NS, FS, HH, 0);
  col_stats_h128<<<HH, 256, 0, stream>>>(A_s, NS, mean, inv);
  bn_relu_inplace<<<GRD, TB, 0, stream>>>(A_s, (size_t)NS * HH, mean, inv, g_in + 2 * HH, bt_in + 2 * HH);

  // ---- 3 hetero layers ----
  for (int L = 0; L < 3; ++L) {
    const float* WlL = Wl + (size_t)L * 6 * HH * HH;
    const float* WrL = Wr + (size_t)L * 6 * HH * HH;
    const float* blL = bl + (size_t)L * 6 * HH;
    auto Wle = [&](int e) { return WlL + (size_t)e * HH * HH; };
    auto Wre = [&](int e) { return WrL + (size_t)e * HH * HH; };
    auto ble = [&](int e) { return blL + (size_t)e * HH; };

    // --- o_d (dst=drug): edges gd(1, src=gene), sd(5, src=dis) ---
    fill_f32<<<GRD, TB, 0, stream>>>(o_d, 0.f, (size_t)ND * HH);
    fill_f32<<<GRD, TB, 0, stream>>>(sb, 0.f, (size_t)ND * HH);
    scatter_add_h128<<<(E_gd + 7) / 8, 256, 0, stream>>>(A_g, src_gd, dst_gd, E_gd, sb);
    gemm_launch(stream, sb, Wle(1), nullptr, r_gd, o_d, ND, HH, HH, 1);
    fill_f32<<<GRD, TB, 0, stream>>>(sb, 0.f, (size_t)ND * HH);
    scatter_add_h128<<<(E_sd + 7) / 8, 256, 0, stream>>>(A_s, src_sd, dst_sd, E_sd, sb);
    gemm_launch(stream, sb, Wle(5), nullptr, r_sd, o_d, ND, HH, HH, 1);
    gemm_launch(stream, A_d, Wre(1), nullptr, nullptr, o_d, ND, HH, HH, 1);
    gemm_launch(stream, A_d, Wre(5), nullptr, nullptr, o_d, ND, HH, HH, 1);

    // --- o_g (dst=gene): edges dg(0, src=drug), sg(3, src=dis) ---
    fill_f32<<<GRD, TB, 0, stream>>>(o_g, 0.f, (size_t)NG * HH);
    fill_f32<<<GRD, TB, 0, stream>>>(sb, 0.f, (size_t)NG * HH);
    scatter_add_h128<<<(E_dg + 7) / 8, 256, 0, stream>>>(A_d, src_dg, dst_dg, E_dg, sb);
    gemm_launch(stream, sb, Wle(0), nullptr, r_dg, o_g, NG, HH, HH, 1);
    fill_f32<<<GRD, TB, 0, stream>>>(sb, 0.f, (size_t)NG * HH);
    scatter_add_h128<<<(E_sg + 7) / 8, 256, 0, stream>>>(A_s, src_sg, dst_sg, E_sg, sb);
    gemm_launch(stream, sb, Wle(3), nullptr, r_sg, o_g, NG, HH, HH, 1);
    gemm_launch(stream, A_g, Wre(0), nullptr, nullptr, o_g, NG, HH, HH, 1);
    gemm_launch(stream, A_g, Wre(3), nullptr, nullptr, o_g, NG, HH, HH, 1);

    // --- o_s (dst=dis): edges gs(2, src=gene), ds(4, src=drug) ---
    fill_f32<<<GRD, TB, 0, stream>>>(o_s, 0.f, (size_t)NS * HH);
    fill_f32<<<GRD, TB, 0, stream>>>(sb, 0.f, (size_t)NS * HH);
    scatter_add_h128<<<(E_gs + 7) / 8, 256, 0, stream>>>(A_g, src_gs, dst_gs, E_gs, sb);
    gemm_launch(stream, sb, Wle(2), nullptr, r_gs, o_s, NS, HH, HH, 1);
    fill_f32<<<GRD, TB, 0, stream>>>(sb, 0.f, (size_t)NS * HH);
    scatter_add_h128<<<(E_ds + 7) / 8, 256, 0, stream>>>(A_d, src_ds, dst_ds, E_ds, sb);
    gemm_launch(stream, sb, Wle(4), nullptr, r_ds, o_s, NS, HH, HH, 1);
    gemm_launch(stream, A_s, Wre(2), nullptr, nullptr, o_s, NS, HH, HH, 1);
    gemm_launch(stream, A_s, Wre(4), nullptr, nullptr, o_s, NS, HH, HH, 1);

    // --- finish (0.5*(.)+biases), BN, ReLU, residual — after ALL o computed ---
    finish_o<<<GRD, TB, 0, stream>>>(o_d, ble(1), ble(5), (size_t)ND * HH);
    col_stats_h128<<<HH, 256, 0, stream>>>(o_d, ND, mean, inv);
    bn_relu_residual<<<GRD, TB, 0, stream>>>(o_d, A_d, (size_t)ND * HH, mean, inv,
                                             g_conv + (size_t)(L * 3 + 0) * HH, bt_conv + (size_t)(L * 3 + 0) * HH);

    finish_o<<<GRD, TB, 0, stream>>>(o_g, ble(0), ble(3), (size_t)NG * HH);
    col_stats_h128<<<HH, 256, 0, stream>>>(o_g, NG, mean, inv);
    bn_relu_residual<<<GRD, TB, 0, stream>>>(o_g, A_g, (size_t)NG * HH, mean, inv,
                                             g_conv + (size_t)(L * 3 + 1) * HH, bt_conv + (size_t)(L * 3 + 1) * HH);

    finish_o<<<GRD, TB, 0, stream>>>(o_s, ble(2), ble(4), (size_t)NS * HH);
    col_stats_h128<<<HH, 256, 0, stream>>>(o_s, NS, mean, inv);
    bn_relu_residual<<<GRD, TB, 0, stream>>>(o_s, A_s, (size_t)NS * HH, mean, inv,
                                             g_conv + (size_t)(L * 3 + 2) * HH, bt_conv + (size_t)(L * 3 + 2) * HH);
  }

  // ---- final projection H -> O, concatenated outputs ----
  float* out = (float*)d_out;
  gemm_launch(stream, A_d, W_fin, b_fin, nullptr, out, ND, HH, OO, 0);
  gemm_launch(stream, A_g, W_fin, b_fin, nullptr, out + (size_t)ND * OO, NG, HH, OO, 0);
  gemm_launch(stream, A_s, W_fin, b_fin, nullptr, out + (size_t)(ND + NG) * OO, NS, HH, OO, 0);
}